// Embedding_61366492725854
// MI455X (gfx1250) — compile-verified
//
#include <hip/hip_runtime.h>

// CDNA5 / gfx1250 fp32 GEMM: C[M,128] = A[M,128] x B[128,128]
// M = 64*4096 = 262144.  Memory-bound (AI ~32 FLOP/B), so keep fp32 and use
// V_WMMA_F32_16X16X4_F32 (exact precision vs. reference einsum).

typedef __attribute__((ext_vector_type(2))) float v2f;
typedef __attribute__((ext_vector_type(8))) float v8f;

constexpr int K_DIM = 128;   // I (contraction)
constexpr int N_DIM = 128;   // E
constexpr int M_TOTAL = 64 * 4096;

constexpr int WAVES_PER_BLOCK  = 8;    // 256 threads = 8 wave32
constexpr int STRIPS_PER_WAVE  = 4;    // each wave does 4 x (16 rows x 128 cols)
constexpr int THREADS          = WAVES_PER_BLOCK * 32;
constexpr int STRIPS_PER_BLOCK = WAVES_PER_BLOCK * STRIPS_PER_WAVE;   // 32
constexpr int NUM_BLOCKS       = M_TOTAL / (16 * STRIPS_PER_BLOCK);   // 512

// LDS holds B swizzled: element (k, e) lives at dword  e*128 + ((k + 4*(e&15)) & 127).
// Load side: lane l (e = nn*16 + (l&15)) reads v2f at k' = (kk*4 + 2*hi + 4*(l&15)) & 127,
// which is even => the (k,k+1) pair never crosses the &127 wrap, and banks
// (k' mod 64) are all-distinct across the 32 lanes (stride-4 rotation) => conflict-free.
__device__ __forceinline__ int b_swz(int k, int e) {
    return e * K_DIM + ((k + 4 * (e & 15)) & (K_DIM - 1));
}

__global__ __launch_bounds__(THREADS)
void embed_gemm_f32_wmma(const float* __restrict__ A,
                         const float* __restrict__ Bm,
                         float* __restrict__ C) {
    __shared__ float ldsB[K_DIM * N_DIM];   // exactly 64 KB

    const int tid = threadIdx.x;

    // Cooperative, coalesced load of the 128x128 embedding into swizzled LDS.
    #pragma unroll
    for (int i = tid; i < K_DIM * N_DIM; i += THREADS) {
        const int k = i >> 7;        // row of B (contraction index)
        const int e = i & (N_DIM - 1);
        ldsB[b_swz(k, e)] = Bm[i];
    }
    __syncthreads();

    const int wave = tid >> 5;
    const int lane = tid & 31;
    const int lm   = lane & 15;      // row within 16x16 tile / column within N-tile
    const int hi   = lane >> 4;      // selects K+2 half per WMMA f32 A/B layout

    for (int it = 0; it < STRIPS_PER_WAVE; ++it) {
        const int strip = (blockIdx.x * WAVES_PER_BLOCK + wave) * STRIPS_PER_WAVE + it;
        const int m0 = strip * 16;

        // Lane's A row pointer: row m0+lm, starting at K offset 2*hi.
        // Per K-step kk this lane needs floats (kk*4 + 2*hi, +1): one b64 load.
        const float* aRow = A + (size_t)(m0 + lm) * K_DIM + 2 * hi;

        // Prefetch this wave's next strip of A (gfx1250 global_prefetch_b8).
        if (it + 1 < STRIPS_PER_WAVE)
            __builtin_prefetch(aRow + 16 * K_DIM, 0, 0);

        v8f acc[8];
        #pragma unroll
        for (int nn = 0; nn < 8; ++nn)
            #pragma unroll
            for (int r = 0; r < 8; ++r)
                acc[nn][r] = 0.0f;

        #pragma unroll 4
        for (int kk = 0; kk < K_DIM / 4; ++kk) {
            const v2f a = *(const v2f*)(aRow + kk * 4);
            const int kBase = (kk * 4 + 2 * hi + 4 * lm) & (K_DIM - 1); // swizzled k'
            #pragma unroll
            for (int nn = 0; nn < 8; ++nn) {
                const int e = nn * 16 + lm;
                const v2f b = *(const v2f*)(ldsB + e * K_DIM + kBase);
                acc[nn] = __builtin_amdgcn_wmma_f32_16x16x4_f32(
                    /*neg_a=*/false, a, /*neg_b=*/false, b,
                    /*c_mod=*/(short)0, acc[nn],
                    /*reuse_a=*/false, /*reuse_b=*/false);
            }
        }

        // Store C strip: VGPR r of a 16x16 f32 D tile holds rows r (lanes 0-15)
        // and r+8 (lanes 16-31), column = lane&15.
        float* cBase = C + (size_t)m0 * N_DIM;
        #pragma unroll
        for (int nn = 0; nn < 8; ++nn) {
            #pragma unroll
            for (int r = 0; r < 8; ++r) {
                const int m = r + 8 * hi;
                cBase[(size_t)m * N_DIM + nn * 16 + lm] = acc[nn][r];
            }
        }
    }
}

extern "C" void kernel_launch(void* const* d_in, const int* in_sizes, int n_in,
                              void* d_out, int out_size, void* d_ws, size_t ws_size,
                              hipStream_t stream) {
    const float* A  = (const float*)d_in[0];   // (64, 4096, 128) fp32
    const float* Bm = (const float*)d_in[1];   // (128, 128) fp32
    float* C        = (float*)d_out;           // (64, 4096, 128) fp32

    embed_gemm_f32_wmma<<<dim3(NUM_BLOCKS), dim3(THREADS), 0, stream>>>(A, Bm, C);
}